// SelectiveSSM_56856777064461
// MI455X (gfx1250) — compile-verified
//
#include <hip/hip_runtime.h>
#include <math.h>
#include <stdint.h>

typedef __attribute__((ext_vector_type(2))) float v2f;
typedef __attribute__((ext_vector_type(8))) float v8f;
typedef __attribute__((ext_vector_type(4))) unsigned int v4u;
typedef __attribute__((ext_vector_type(4))) int v4i;
typedef __attribute__((ext_vector_type(8))) int v8i;

#define BB 8
#define LL 2048
#define DD 1024
#define NN 16
#define MM (BB * LL)

#define KC   32      // K chunk per LDS stage (dwords per tile row)
#define XPAD 36      // effective LDS row stride in floats (KC + 4 pad dwords)

__device__ __forceinline__ float softplus_f(float s) {
    // matches jax.nn.softplus = max(s,0) + log1p(exp(-|s|))
    return fmaxf(s, 0.0f) + log1pf(__expf(-fabsf(s)));
}

// ---------------------------------------------------------------------------
// TDM: issue a 2D tile load (tile_d0 x tile_d1 elements of 4B) from a row-major
// tensor into LDS, with 4-dword padding inserted after every 32 dwords so rows
// land at stride XPAD=36 floats (bank-conflict-free for the WMMA frag reads).
// D# layout per CDNA5 ISA 8.3/8.4.
// ---------------------------------------------------------------------------
__device__ __forceinline__ void tdm_load_tile_2d(
    unsigned int lds_byte_off, const float* gptr,
    unsigned int tensor_d0, unsigned int tensor_d1,
    unsigned int tile_d0, unsigned int tile_d1,
    unsigned int stride0_elems)
{
    const unsigned long long ga = (unsigned long long)(uintptr_t)gptr;

    v4u g0;
    g0.x = 1u;                                          // count=1, user mode
    g0.y = lds_byte_off;                                // lds_addr
    g0.z = (unsigned int)(ga & 0xFFFFFFFFull);          // global_addr[31:0]
    g0.w = (unsigned int)((ga >> 32) & 0x01FFFFFFull)   // global_addr[56:32]
         | (2u << 30);                                  // type=2 ("image")

    v8i g1;
    g1[0] = (int)((2u << 16)       // data_size = 2 -> 4 bytes
                | (1u << 20)       // pad_enable
                | (4u << 22)       // pad_interval: 2^(4+1)=32 dwords
                | (3u << 25));     // pad_amount: 3+1 = 4 dwords
    g1[1] = (int)((tensor_d0 & 0xFFFFu) << 16);                    // dim0[15:0]
    g1[2] = (int)((tensor_d0 >> 16) | ((tensor_d1 & 0xFFFFu) << 16)); // dim0[31:16]|dim1[15:0]
    g1[3] = (int)((tensor_d1 >> 16) | (tile_d0 << 16));            // dim1[31:16]|tile_dim0
    g1[4] = (int)(tile_d1 & 0xFFFFu);                              // tile_dim1 (tile_dim2=0)
    g1[5] = (int)stride0_elems;                                    // tensor_dim0_stride[31:0]
    g1[6] = 0;                                                     // stride0[47:32]|stride1[15:0]
    g1[7] = 0;

    v4i g2 = {0, 0, 0, 0};
    v4i g3 = {0, 0, 0, 0};
#if __clang_major__ >= 23
    v8i g4 = {0, 0, 0, 0, 0, 0, 0, 0};
    __builtin_amdgcn_tensor_load_to_lds(g0, g1, g2, g3, g4, 0);
#else
    __builtin_amdgcn_tensor_load_to_lds(g0, g1, g2, g3, 0);
#endif
}

// ---------------------------------------------------------------------------
// Kernel 1: delta = softplus(X @ W_delta^T + b_delta), X:(M,D) W:(D,D)
// 256 threads (8 waves), tile 64(M) x 64(E). TDM double-buffered K staging:
// wave 0 DMAs chunk i+1 while all waves run WMMA on chunk i.
// ---------------------------------------------------------------------------
__global__ __launch_bounds__(256) void delta_gemm_kernel(
    const float* __restrict__ X, const float* __restrict__ W,
    const float* __restrict__ bias, float* __restrict__ out)
{
    __shared__ float Xs[2][64 * XPAD];
    __shared__ float Ws[2][64 * XPAD];

    const int m0   = blockIdx.x * 64;
    const int e0   = blockIdx.y * 64;
    const int tid  = threadIdx.x;
    const int wave = tid >> 5;
    const int lane = tid & 31;
    const int l16  = lane & 15;
    const int hi   = lane >> 4;          // selects K+{0,1} vs K+{2,3}
    const int sm   = (wave & 3) * 16;    // M sub-tile
    const int se   = (wave >> 2) * 32;   // E sub-tile pair base

    v8f acc0 = {};
    v8f acc1 = {};

    const unsigned int xs0 = (unsigned int)(uintptr_t)&Xs[0][0];
    const unsigned int xs1 = (unsigned int)(uintptr_t)&Xs[1][0];
    const unsigned int ws0 = (unsigned int)(uintptr_t)&Ws[0][0];
    const unsigned int ws1 = (unsigned int)(uintptr_t)&Ws[1][0];

    const int nch = DD / KC;             // 32 chunks

    if (wave == 0) {
        tdm_load_tile_2d(xs0, X + (size_t)m0 * DD, DD, MM, KC, 64, DD);
        tdm_load_tile_2d(ws0, W + (size_t)e0 * DD, DD, DD, KC, 64, DD);
    }

    for (int i = 0; i < nch; ++i) {
        const int cur = i & 1;
        if (wave == 0) {
            if (i + 1 < nch) {
                const int kc = (i + 1) * KC;
                tdm_load_tile_2d(cur ? xs0 : xs1,
                                 X + (size_t)m0 * DD + kc, DD, MM, KC, 64, DD);
                tdm_load_tile_2d(cur ? ws0 : ws1,
                                 W + (size_t)e0 * DD + kc, DD, DD, KC, 64, DD);
                __builtin_amdgcn_s_wait_tensorcnt(2);   // chunk i landed
            } else {
                __builtin_amdgcn_s_wait_tensorcnt(0);   // drain
            }
        }
        __syncthreads();

        const float* xs = &Xs[cur][0];
        const float* wsp = &Ws[cur][0];
        #pragma unroll
        for (int k = 0; k < KC; k += 4) {
            const int kk = k + 2 * hi;
            v2f a  = *(const v2f*)&xs[(sm + l16) * XPAD + kk];
            v2f b0 = *(const v2f*)&wsp[(se + l16) * XPAD + kk];
            v2f b1 = *(const v2f*)&wsp[(se + 16 + l16) * XPAD + kk];
            acc0 = __builtin_amdgcn_wmma_f32_16x16x4_f32(
                false, a, false, b0, (short)0, acc0, false, false);
            acc1 = __builtin_amdgcn_wmma_f32_16x16x4_f32(
                false, a, false, b1, (short)0, acc1, false, false);
        }
        __syncthreads();   // everyone done with buf cur before it is re-DMAed
    }

    const int ee0 = e0 + se + l16;
    const int ee1 = ee0 + 16;
    const float bia0 = bias[ee0];
    const float bia1 = bias[ee1];
    #pragma unroll
    for (int v = 0; v < 8; ++v) {
        const int m = m0 + sm + v + 8 * hi;  // C/D layout: VGPR v -> row v + 8*hi
        out[(size_t)m * DD + ee0] = softplus_f(acc0[v] + bia0);
        out[(size_t)m * DD + ee1] = softplus_f(acc1[v] + bia1);
    }
}

// ---------------------------------------------------------------------------
// Kernel 2: B_in = X @ W_B^T, C_in = X @ W_C^T  (both M x 16)
// 256 threads (8 waves), 128 rows/WG; wave w owns rows m0+16w, accumulates
// both the B and C 16x16 tiles.
// ---------------------------------------------------------------------------
__global__ __launch_bounds__(256) void bc_gemm_kernel(
    const float* __restrict__ X, const float* __restrict__ WB,
    const float* __restrict__ WC, float* __restrict__ Bout,
    float* __restrict__ Cout)
{
    __shared__ float Xs[128 * XPAD];
    __shared__ float WbS[16 * XPAD];
    __shared__ float WcS[16 * XPAD];

    const int m0   = blockIdx.x * 128;
    const int tid  = threadIdx.x;
    const int wave = tid >> 5;
    const int lane = tid & 31;
    const int l16  = lane & 15;
    const int hi   = lane >> 4;

    v8f accB = {};
    v8f accC = {};

    for (int kc = 0; kc < DD; kc += KC) {
        #pragma unroll
        for (int i = 0; i < 4; ++i) {
            const int idx = tid + 256 * i;
            const int r = idx >> 3;
            const int c = (idx & 7) << 2;
            *(float4*)&Xs[r * XPAD + c] =
                *(const float4*)(X + (size_t)(m0 + r) * DD + kc + c);
        }
        {
            const int t2 = tid & 127;
            const int r = t2 >> 3;
            const int c = (t2 & 7) << 2;
            const float* src = (tid < 128) ? WB : WC;
            float* dst       = (tid < 128) ? WbS : WcS;
            *(float4*)&dst[r * XPAD + c] =
                *(const float4*)(src + (size_t)r * DD + kc + c);
        }
        __syncthreads();

        #pragma unroll
        for (int k = 0; k < KC; k += 4) {
            const int kk = k + 2 * hi;
            v2f a  = *(const v2f*)&Xs[(wave * 16 + l16) * XPAD + kk];
            v2f bB = *(const v2f*)&WbS[l16 * XPAD + kk];
            v2f bC = *(const v2f*)&WcS[l16 * XPAD + kk];
            accB = __builtin_amdgcn_wmma_f32_16x16x4_f32(
                false, a, false, bB, (short)0, accB, false, false);
            accC = __builtin_amdgcn_wmma_f32_16x16x4_f32(
                false, a, false, bC, (short)0, accC, false, false);
        }
        __syncthreads();
    }

    #pragma unroll
    for (int v = 0; v < 8; ++v) {
        const int m = m0 + wave * 16 + v + 8 * hi;
        Bout[(size_t)m * NN + l16] = accB[v];
        Cout[(size_t)m * NN + l16] = accC[v];
    }
}

// ---------------------------------------------------------------------------
// Kernel 3: sequential selective scan over L. One thread per (b, d); h[16] in
// registers. Y holds delta on input (kernel 1) and y on output (same-thread
// read-then-write per address). Streaming reads are prefetched 8 steps ahead.
// ---------------------------------------------------------------------------
__global__ __launch_bounds__(128) void scan_kernel(
    const float* __restrict__ X, float* __restrict__ Y,
    const float* __restrict__ Bin, const float* __restrict__ Cin,
    const float* __restrict__ A_log, const float* __restrict__ Dp,
    float* __restrict__ Hf)
{
    __shared__ float bsh[NN];
    __shared__ float csh[NN];

    const int tid = threadIdx.x;
    const int d   = blockIdx.x * 128 + tid;
    const int b   = blockIdx.y;

    float A[NN];
    float h[NN];
    #pragma unroll
    for (int n = 0; n < NN; ++n) {
        A[n] = -__expf(A_log[(size_t)d * NN + n]);
        h[n] = 0.0f;
    }
    const float dpar = Dp[d];

    const size_t base   = (size_t)b * LL * DD + d;
    const size_t bcbase = (size_t)b * LL * NN;

    for (int t = 0; t < LL; ++t) {
        if (tid < NN)
            bsh[tid] = Bin[bcbase + (size_t)t * NN + tid];
        else if (tid < 2 * NN)
            csh[tid - NN] = Cin[bcbase + (size_t)t * NN + (tid - NN)];
        __syncthreads();

        const size_t idx = base + (size_t)t * DD;
        if (t + 8 < LL) {
            __builtin_prefetch(&Y[idx + 8 * DD], 0, 0);
            __builtin_prefetch(&X[idx + 8 * DD], 0, 0);
        }
        const float dv = Y[idx];       // delta (written by kernel 1)
        const float xv = X[idx];
        const float dx = dv * xv;
        float y = dpar * xv;
        #pragma unroll
        for (int n = 0; n < NN; ++n) {
            const float abar = __expf(dv * A[n]);
            h[n] = abar * h[n] + dx * bsh[n];
            y += h[n] * csh[n];
        }
        Y[idx] = y;
        __syncthreads();
    }

    float* hp = Hf + ((size_t)b * DD + d) * NN;
    #pragma unroll
    for (int n = 0; n < NN; ++n) hp[n] = h[n];
}

// ---------------------------------------------------------------------------
extern "C" void kernel_launch(void* const* d_in, const int* in_sizes, int n_in,
                              void* d_out, int out_size, void* d_ws, size_t ws_size,
                              hipStream_t stream)
{
    const float* x       = (const float*)d_in[0];  // (B, L, D)
    const float* A_log   = (const float*)d_in[1];  // (D, N)
    const float* D_param = (const float*)d_in[2];  // (D,)
    const float* W_B     = (const float*)d_in[3];  // (N, D)
    const float* W_C     = (const float*)d_in[4];  // (N, D)
    const float* W_delta = (const float*)d_in[5];  // (D, D)
    const float* b_delta = (const float*)d_in[6];  // (D,)

    float* y  = (float*)d_out;                     // (B, L, D): delta then y
    float* hf = y + (size_t)MM * DD;               // (B, D, N)

    float* Bin = (float*)d_ws;                     // (M, N)
    float* Cin = Bin + (size_t)MM * NN;            // (M, N)

    delta_gemm_kernel<<<dim3(MM / 64, DD / 64), 256, 0, stream>>>(
        x, W_delta, b_delta, y);
    bc_gemm_kernel<<<dim3(MM / 128), 256, 0, stream>>>(
        x, W_B, W_C, Bin, Cin);
    scan_kernel<<<dim3(DD / 128, BB), 128, 0, stream>>>(
        x, y, Bin, Cin, A_log, D_param, hf);
}